// SSIM_19516331393405
// MI455X (gfx1250) — compile-verified
//
#include <hip/hip_runtime.h>
#include <stdint.h>

typedef __attribute__((ext_vector_type(2))) float v2f;
typedef __attribute__((ext_vector_type(8))) float v8f;

#define BATCH   32
#define NCH     3
#define HW      512
#define OUTD    502        // 512 - 11 + 1
#define TILE    16
#define NTILES  32         // ceil(502/16)
#define IN_T    26         // TILE + 10
#define XSTRIDE 28         // even row stride -> aligned b64 pair loads
#define TSTRIDE 17

__device__ __forceinline__ v8f wmma_f32(v2f a, v2f b, v8f c) {
  // D = A(16x4) * B(4x16) + C ; fp32 matrix pipe, wave32
  return __builtin_amdgcn_wmma_f32_16x16x4_f32(false, a, false, b, (short)0, c,
                                               false, false);
}

// g[k - n] for the 26x16 band matrix (zero outside band / past K=26), branch-free
__device__ __forceinline__ float wfun(const float* gs, int k, int n) {
  int d = k - n;
  bool in = ((unsigned)d < 11u) && (k < 26);
  float g = gs[in ? d : 0];
  return in ? g : 0.0f;
}

__global__ void __launch_bounds__(32)
ssim_wmma_kernel(const float* __restrict__ img1, const float* __restrict__ img2,
                 float* __restrict__ out) {
  __shared__ float gs[16];
  __shared__ float x1s[IN_T][XSTRIDE];
  __shared__ float x2s[IN_T][XSTRIDE];
  __shared__ float Ts[32][TSTRIDE];

  const int lane = threadIdx.x;
  int t = blockIdx.x;
  const int tx = t & 31; t >>= 5;
  const int ty = t & 31; t >>= 5;
  const int ch = t % NCH;
  const int b  = t / NCH;
  const int y0 = ty * TILE, x0 = tx * TILE;

  // 11-tap Gaussian, sigma=1.5, normalized (raw sum = 3.7592318)
  if (lane < 11) {
    float d = (float)(lane - 5);
    gs[lane] = __expf(-d * d * (1.0f / 4.5f)) * (1.0f / 3.7592318f);
  }

  // ---- async-fill 26x26 halo tiles of both images straight into LDS ----
  // Edge tiles: clamp to the last row/col. Clamped (finite) garbage only feeds
  // output pixels >= 502 which are masked out of the reduction, and all padded
  // K positions meet an exactly-zero Gaussian weight, so no masking is needed.
  const size_t plane_off = ((size_t)(b * NCH + ch)) * HW * HW;
  const float* p1 = img1 + plane_off;
  const float* p2 = img2 + plane_off;
  for (int i = lane; i < IN_T * IN_T; i += 32) {
    int r = i / IN_T;
    int c = i - r * IN_T;
    int gy = y0 + r; gy = gy < HW ? gy : (HW - 1);
    int gx = x0 + c; gx = gx < HW ? gx : (HW - 1);
    uint32_t voff = (uint32_t)(gy * HW + gx) * 4u;          // byte offset vs base
    uint32_t l1 = (uint32_t)(uintptr_t)&x1s[r][c];          // LDS byte address
    uint32_t l2 = (uint32_t)(uintptr_t)&x2s[r][c];
    asm volatile("global_load_async_to_lds_b32 %0, %1, %2"
                 :: "v"(l1), "v"(voff), "s"(p1) : "memory");
    asm volatile("global_load_async_to_lds_b32 %0, %1, %2"
                 :: "v"(l2), "v"(voff), "s"(p2) : "memory");
  }
  asm volatile("s_wait_asynccnt 0x0" ::: "memory");
  __syncthreads();

  const int N  = lane & 15;   // B/D column, and A row (M)
  const int hi = lane >> 4;   // K sub-offset selector

  // gw[kk] = Gauss band operand; serves as stage-1 B and stage-2 A.
  // Exactly zero outside the band and for k >= 26 (kills all padding).
  v2f gw[7];
#pragma unroll
  for (int kk = 0; kk < 7; ++kk) {
    int kb = 4 * kk + 2 * hi;
    gw[kk].x = wfun(gs, kb, N);
    gw[kk].y = wfun(gs, kb + 1, N);
  }

  const int rA0 = N;                                   // rows 0..15
  const int rowB = 16 + N;                             // rows 16..31
  const int rA1 = rowB < IN_T ? rowB : (IN_T - 1);     // clamp keeps data finite

  // ---- stage 1, K-outer: load each LDS pair once, feed all 5 planes ----
  // acc1[2p+g]: plane p in {x1, x2, x1^2, x2^2, x1*x2}, row group g in {0,1}
  v8f acc1[10];
#pragma unroll
  for (int i = 0; i < 10; ++i) acc1[i] = (v8f){};

#pragma unroll
  for (int kk = 0; kk < 7; ++kk) {
    int kb = 4 * kk + 2 * hi;
    int cc = (kb < 26) ? kb : 24;                      // clamp pad cols (weights=0)
    const v2f u0 = *reinterpret_cast<const v2f*>(&x1s[rA0][cc]);
    const v2f w0 = *reinterpret_cast<const v2f*>(&x2s[rA0][cc]);
    const v2f u1 = *reinterpret_cast<const v2f*>(&x1s[rA1][cc]);
    const v2f w1 = *reinterpret_cast<const v2f*>(&x2s[rA1][cc]);
    acc1[0] = wmma_f32(u0,      gw[kk], acc1[0]);
    acc1[1] = wmma_f32(u1,      gw[kk], acc1[1]);
    acc1[2] = wmma_f32(w0,      gw[kk], acc1[2]);
    acc1[3] = wmma_f32(w1,      gw[kk], acc1[3]);
    acc1[4] = wmma_f32(u0 * u0, gw[kk], acc1[4]);
    acc1[5] = wmma_f32(u1 * u1, gw[kk], acc1[5]);
    acc1[6] = wmma_f32(w0 * w0, gw[kk], acc1[6]);
    acc1[7] = wmma_f32(w1 * w1, gw[kk], acc1[7]);
    acc1[8] = wmma_f32(u0 * w0, gw[kk], acc1[8]);
    acc1[9] = wmma_f32(u1 * w1, gw[kk], acc1[9]);
  }

  // ---- stage 2 per plane: D layout -> LDS T -> Out = Gv^T (16x26) * T ----
  v8f res[5];
#pragma unroll
  for (int p = 0; p < 5; ++p) {
    __syncthreads();
#pragma unroll
    for (int r = 0; r < 8; ++r) {                      // lane holds (M=r+8*hi, N)
      Ts[r + 8 * hi][N]      = acc1[2 * p][r];
      Ts[16 + r + 8 * hi][N] = acc1[2 * p + 1][r];
    }
    __syncthreads();
    v8f acc = (v8f){};
#pragma unroll
    for (int kk = 0; kk < 7; ++kk) {
      int kb = 4 * kk + 2 * hi;
      v2f bb;
      bb.x = Ts[kb][N];                                // rows 26/27 meet weight 0
      bb.y = Ts[kb + 1][N];
      acc = wmma_f32(gw[kk], bb, acc);
    }
    res[p] = acc;
  }

  // ---- SSIM map + masked tile reduction ----
  const float C1c = 1.0e-4f, C2c = 9.0e-4f;
  float sum = 0.0f;
#pragma unroll
  for (int r = 0; r < 8; ++r) {
    int oy = y0 + r + 8 * hi;
    int ox = x0 + N;
    float mu1 = res[0][r], mu2 = res[1][r];
    float e11 = res[2][r], e22 = res[3][r], e12 = res[4][r];
    float mu1sq = mu1 * mu1, mu2sq = mu2 * mu2, mu12 = mu1 * mu2;
    float s1 = e11 - mu1sq, s2 = e22 - mu2sq, s12 = e12 - mu12;
    float v1 = 2.0f * s12 + C2c;
    float v2 = s1 + s2 + C2c;
    float num = (2.0f * mu12 + C1c) * v1;
    float den = (mu1sq + mu2sq + C1c) * v2;
    float ssim = num / den;
    sum += ((oy < OUTD) && (ox < OUTD)) ? ssim : 0.0f;
  }
#pragma unroll
  for (int off = 16; off > 0; off >>= 1)
    sum += __shfl_down(sum, off, 32);
  if (lane == 0)
    atomicAdd(&out[b], sum * (1.0f / ((float)NCH * OUTD * OUTD)));
}

__global__ void ssim_zero_kernel(float* __restrict__ out, int n) {
  int i = blockIdx.x * blockDim.x + threadIdx.x;
  if (i < n) out[i] = 0.0f;
}

extern "C" void kernel_launch(void* const* d_in, const int* in_sizes, int n_in,
                              void* d_out, int out_size, void* d_ws, size_t ws_size,
                              hipStream_t stream) {
  (void)in_sizes; (void)n_in; (void)d_ws; (void)ws_size;
  const float* img1 = (const float*)d_in[0];
  const float* img2 = (const float*)d_in[1];
  float* out = (float*)d_out;

  ssim_zero_kernel<<<1, 64, 0, stream>>>(out, out_size);

  const int nblocks = BATCH * NCH * NTILES * NTILES;  // 98304 tiles, 1 wave each
  ssim_wmma_kernel<<<nblocks, 32, 0, stream>>>(img1, img2, out);
}